// MyNet2_70514773066455
// MI455X (gfx1250) — compile-verified
//
#include <hip/hip_runtime.h>
#include <math.h>

typedef __attribute__((ext_vector_type(2))) float v2f;
typedef __attribute__((ext_vector_type(8))) float v8f;
typedef __attribute__((ext_vector_type(4))) int   v4i;
typedef __attribute__((ext_vector_type(4))) float v4f;

#define NPG 38  // nodes per graph

// -------- Stage 0: agg[i] = x[i]*conv_root + conv_bias (atomic accumulator seed)
__global__ void init_agg_kernel(const float* __restrict__ x,
                                const float* __restrict__ conv_root,
                                const float* __restrict__ conv_bias,
                                float* __restrict__ agg, int n) {
    int i = blockIdx.x * blockDim.x + threadIdx.x;
    if (i < n) agg[i] = fmaf(x[i], conv_root[0], conv_bias[0]);
}

// -------- Stage 1: 60.8M edges: agg[dst] += x[src] * (attr*w + b)
// Edge streams are read once -> non-temporal loads keep x/agg resident in L2.
__global__ void edge_kernel(const int* __restrict__ src, const int* __restrict__ dst,
                            const float* __restrict__ attr, const float* __restrict__ x,
                            const float* __restrict__ ew_p, const float* __restrict__ eb_p,
                            float* __restrict__ agg, long long E) {
    const float w  = ew_p[0];
    const float bb = eb_p[0];
    long long q    = (long long)blockIdx.x * blockDim.x + threadIdx.x;
    long long base = q * 4;
    if (base + 3 < E) {
        v4i s = __builtin_nontemporal_load((const v4i*)src  + q);
        v4i d = __builtin_nontemporal_load((const v4i*)dst  + q);
        v4f a = __builtin_nontemporal_load((const v4f*)attr + q);
        atomicAdd(&agg[d.x], x[s.x] * fmaf(a.x, w, bb));
        atomicAdd(&agg[d.y], x[s.y] * fmaf(a.y, w, bb));
        atomicAdd(&agg[d.z], x[s.z] * fmaf(a.z, w, bb));
        atomicAdd(&agg[d.w], x[s.w] * fmaf(a.w, w, bb));
    } else if (base < E) {
        for (long long e = base; e < E; ++e)
            atomicAdd(&agg[dst[e]], x[src[e]] * fmaf(attr[e], w, bb));
    }
}

// -------- Stage 2: head MLP. One wave32 per 16-graph tile.
// Layer 1 ([16,38]@[38,4]) via fp32 WMMA 16x16x4 (K padded to 40, N padded to 16).
// All fragment loads are unconditional from clamped addresses; padding is done
// with value selects (v_cndmask) so the loop is branch-free straight-line code.
__global__ void head_kernel(const float* __restrict__ nodes,
                            const float* __restrict__ W1, const float* __restrict__ b1,
                            const float* __restrict__ W2, const float* __restrict__ b2,
                            const float* __restrict__ W3, const float* __restrict__ b3,
                            float* __restrict__ out, int num_tiles, int G) {
    __shared__ float z1s[8][16][4];  // per-wave 16x4 layer-1 activations

    const int lane      = threadIdx.x & 31;
    const int waveInBlk = threadIdx.x >> 5;
    const int tile      = blockIdx.x * 8 + waveInBlk;
    if (tile >= num_tiles) return;

    const int  g0       = tile * 16;
    const int  row      = lane & 15;                 // A-frag: matrix row
    const int  col      = lane & 15;                 // B/C/D: matrix column
    const int  khalf    = (lane >> 4) << 1;          // 0 (lanes 0-15) or 2 (16-31)
    const bool rowValid = (g0 + row) < G;
    const int  growc    = rowValid ? (g0 + row) : (G - 1);   // clamped, always legal
    const float* Hrow   = nodes + (long long)growc * NPG;    // 152B-aligned rows
    const bool cok      = col < 4;
    const int  colm     = lane & 3;

    v8f c = {};  // 16x16 fp32 accumulator (cols 0-3 meaningful)
#pragma unroll
    for (int k0 = 0; k0 < 40; k0 += 4) {
        const int  kk   = k0 + khalf;                // even: 0..38
        const int  kc   = (kk <= 36) ? kk : 36;      // clamped, always legal
        const bool k0ok = kk < NPG;                  // false only for kk==38
        const bool k1ok = (kk + 1) < NPG;

        // A[m][k],A[m][k+1]: one aligned 8B load (kc even, row base 8B aligned)
        const v2f av = *(const v2f*)(Hrow + kc);
        // B[k][n],B[k+1][n] for n<4: two unconditional scalar loads
        const float bv0 = W1[kc * 4 + colm];
        const float bv1 = W1[(kc + 1) * 4 + colm];

        v2f a, b;
        a.x = (rowValid && k0ok) ? av.x : 0.0f;
        a.y = (rowValid && k1ok) ? av.y : 0.0f;
        b.x = (cok && k0ok) ? bv0 : 0.0f;
        b.y = (cok && k1ok) ? bv1 : 0.0f;

        // D = A x B + C   (v_wmma_f32_16x16x4_f32)
        c = __builtin_amdgcn_wmma_f32_16x16x4_f32(
                /*neg_a=*/false, a, /*neg_b=*/false, b,
                /*c_mod=*/(short)0, c, /*reuse_a=*/false, /*reuse_b=*/false);
    }

    // bias + relu; C/D layout: vgpr v, lanes 0-15 -> M=v, lanes 16-31 -> M=v+8
    if (cok) {
        const float bc    = b1[col];
        const int   rbase = (lane >> 4) ? 8 : 0;
#pragma unroll
        for (int v = 0; v < 8; ++v) {
            float z = c[v] + bc;
            z1s[waveInBlk][rbase + v][col] = z > 0.0f ? z : 0.0f;
        }
    }
    __syncthreads();

    // Layers 2 and 3: one graph per lane (lanes 0-15)
    if (lane < 16 && (g0 + lane) < G) {
        float z0 = z1s[waveInBlk][lane][0];
        float z1 = z1s[waveInBlk][lane][1];
        float z2 = z1s[waveInBlk][lane][2];
        float z3 = z1s[waveInBlk][lane][3];
        float acc = 0.0f;
#pragma unroll
        for (int j = 0; j < 4; ++j) {
            float h = fmaf(z0, W2[0 * 4 + j],
                      fmaf(z1, W2[1 * 4 + j],
                      fmaf(z2, W2[2 * 4 + j],
                      fmaf(z3, W2[3 * 4 + j], b2[j]))));
            h = h > 0.0f ? h : 0.0f;
            acc = fmaf(h, W3[j], acc);
        }
        acc += b3[0];
        out[g0 + lane] = 11.0f / (1.0f + expf(-acc));  // 11*sigmoid
    }
}

extern "C" void kernel_launch(void* const* d_in, const int* in_sizes, int n_in,
                              void* d_out, int out_size, void* d_ws, size_t ws_size,
                              hipStream_t stream) {
    const float* x         = (const float*)d_in[0];
    const int*   eidx      = (const int*)  d_in[1];
    const float* eattr     = (const float*)d_in[2];
    const float* conv_root = (const float*)d_in[3];
    const float* conv_bias = (const float*)d_in[4];
    const float* enw       = (const float*)d_in[5];
    const float* enb       = (const float*)d_in[6];
    const float* W1        = (const float*)d_in[7];
    const float* b1        = (const float*)d_in[8];
    const float* W2        = (const float*)d_in[9];
    const float* b2        = (const float*)d_in[10];
    const float* W3        = (const float*)d_in[11];
    const float* b3        = (const float*)d_in[12];
    float*       out       = (float*)d_out;

    const int       N   = in_sizes[0];          // 3,800,000 nodes
    const long long E   = in_sizes[2];          // 60,800,000 edges
    const int       G   = N / NPG;              // 100,000 graphs
    const int*      src = eidx;                 // edge_index[0]
    const int*      dst = eidx + E;             // edge_index[1]
    float*          agg = (float*)d_ws;         // N floats of scratch (15.2 MB)

    // Stage 0: seed accumulator with root/bias term
    init_agg_kernel<<<(N + 255) / 256, 256, 0, stream>>>(x, conv_root, conv_bias, agg, N);

    // Stage 1: edge gather/scale/scatter-add (bandwidth-bound, ~730 MB stream)
    const long long quads = (E + 3) / 4;
    edge_kernel<<<(unsigned)((quads + 255) / 256), 256, 0, stream>>>(
        src, dst, eattr, x, enw, enb, agg, E);

    // Stage 2: per-graph MLP head with fp32 WMMA
    const int num_tiles = (G + 15) / 16;        // 6250 wave-tiles
    head_kernel<<<(num_tiles + 7) / 8, 256, 0, stream>>>(
        agg, W1, b1, W2, b2, W3, b3, out, num_tiles, G);
}